// ListNetLoss_73057393705013
// MI455X (gfx1250) — compile-verified
//
#include <hip/hip_runtime.h>
#include <stdint.h>

// ListNet KL loss: bandwidth-bound masked double-log-softmax reduction.
// CDNA5 path: per-row tile DMA via Tensor Data Mover (tensor_load_to_lds),
// s_wait_tensorcnt, wave32 shfl_xor reductions. No matmul -> no WMMA by design.

#define B_ROWS 8192
#define L_COLS 4096
#define TPB    256
#define NWAVES (TPB / 32)

typedef unsigned int uint32x4 __attribute__((ext_vector_type(4)));
typedef int          int32x8  __attribute__((ext_vector_type(8)));
typedef int          int32x4  __attribute__((ext_vector_type(4)));

__device__ __forceinline__ void tdm_load_row(uint64_t gaddr, unsigned int lds_off) {
    // D# group 0 (128b): [1:0]=count=1, [63:32]=lds_addr,
    // [120:64]=global_addr (57b), [127:126]=type=2.
    uint32x4 g0;
    g0[0] = 1u;                                            // count=1, gather off
    g0[1] = lds_off;                                       // LDS byte address
    g0[2] = (unsigned int)(gaddr & 0xFFFFFFFFull);         // global_addr[31:0]
    g0[3] = (unsigned int)((gaddr >> 32) & 0x1FFFFFFull)   // global_addr[56:32]
          | 0x80000000u;                                   // type=2 ("image")
    // D# group 1 (256b): data_size=4B(code 2) @ [17:16];
    // tensor_dim0=4096 @ [79:48]; tensor_dim1=1 @ [111:80];
    // tile_dim0=4096 @ [127:112]; tile_dim1=1 @ [143:128];
    // tensor_dim0_stride=4096 @ [207:160].
    int32x8 g1;
    g1[0] = 0x00020000;                // workgroup_mask=0, data_size=2 (4B)
    g1[1] = (int)(L_COLS << 16);       // tensor_dim0[15:0] in bits 31:16
    g1[2] = (int)(1u << 16);           // tensor_dim0[31:16]=0 | tensor_dim1[15:0]=1
    g1[3] = (int)(L_COLS << 16);       // tensor_dim1[31:16]=0 | tile_dim0=4096
    g1[4] = 1;                         // tile_dim1=1, tile_dim2=0
    g1[5] = L_COLS;                    // tensor_dim0_stride[31:0]
    g1[6] = 0;                         // stride hi / dim1_stride (unused: 1 line)
    g1[7] = 0;
    int32x4 gz4 = {0, 0, 0, 0};        // groups 2/3 unused (<=2D tile)
    int32x8 gz8 = {0, 0, 0, 0, 0, 0, 0, 0};
    __builtin_amdgcn_tensor_load_to_lds(g0, g1, gz4, gz4, gz8, 0);
}

__global__ __launch_bounds__(TPB) void listnet_row_kernel(
    const float* __restrict__ pred, const float* __restrict__ targ,
    float* __restrict__ row_kl, float* __restrict__ row_valid)
{
    __shared__ __align__(16) float ldsP[L_COLS];
    __shared__ __align__(16) float ldsT[L_COLS];
    __shared__ float redA[NWAVES];
    __shared__ float redB[NWAVES];
    __shared__ float redC[NWAVES];

    const int row  = blockIdx.x;
    const int tid  = threadIdx.x;
    const int lane = tid & 31;
    const int wav  = tid >> 5;

    // ---- TDM: DMA this row of each matrix into LDS (wave 0 issues) ----
    if (tid == 0) {
        uint64_t gp = (uint64_t)(uintptr_t)(pred + (size_t)row * L_COLS);
        uint64_t gt = (uint64_t)(uintptr_t)(targ + (size_t)row * L_COLS);
        unsigned int lp = (unsigned int)(uintptr_t)(void*)ldsP; // low 32 = LDS offset
        unsigned int lt = (unsigned int)(uintptr_t)(void*)ldsT;
        tdm_load_row(gp, lp);
        tdm_load_row(gt, lt);
        __builtin_amdgcn_s_wait_tensorcnt(0);
    }
    __syncthreads();

    // ---- Stage row slice into registers (one LDS read per element) ----
    float p[16], t[16], et[16];
#pragma unroll
    for (int k = 0; k < 4; ++k) {
        float4 pv = reinterpret_cast<const float4*>(ldsP)[tid + k * TPB];
        float4 tv = reinterpret_cast<const float4*>(ldsT)[tid + k * TPB];
        p[4*k+0] = pv.x; p[4*k+1] = pv.y; p[4*k+2] = pv.z; p[4*k+3] = pv.w;
        t[4*k+0] = tv.x; t[4*k+1] = tv.y; t[4*k+2] = tv.z; t[4*k+3] = tv.w;
    }

    // ---- Pass A: masked row max (both matrices) + valid count ----
    float maxP = -3.0e38f, maxT = -3.0e38f;
    float cnt = 0.0f;
#pragma unroll
    for (int j = 0; j < 16; ++j) {
        bool valid = (t[j] == t[j]);               // !isnan(target)
        maxP = valid ? fmaxf(maxP, p[j]) : maxP;
        maxT = valid ? fmaxf(maxT, t[j]) : maxT;
        cnt += valid ? 1.0f : 0.0f;
    }
#pragma unroll
    for (int off = 16; off > 0; off >>= 1) {
        maxP = fmaxf(maxP, __shfl_xor(maxP, off, 32));
        maxT = fmaxf(maxT, __shfl_xor(maxT, off, 32));
        cnt += __shfl_xor(cnt, off, 32);
    }
    if (lane == 0) { redA[wav] = maxP; redB[wav] = maxT; redC[wav] = cnt; }
    __syncthreads();
    float mP = -3.0e38f, mT = -3.0e38f, cntF = 0.0f;
#pragma unroll
    for (int w = 0; w < NWAVES; ++w) {
        mP = fmaxf(mP, redA[w]); mT = fmaxf(mT, redB[w]); cntF += redC[w];
    }
    __syncthreads();

    // ---- Pass B: masked sum of exponentials (cache exp(t - mT)) ----
    float sP = 0.0f, sT = 0.0f;
#pragma unroll
    for (int j = 0; j < 16; ++j) {
        bool valid = (t[j] == t[j]);
        float ep = valid ? __expf(p[j] - mP) : 0.0f;
        float e  = valid ? __expf(t[j] - mT) : 0.0f;
        et[j] = e;
        sP += ep;
        sT += e;
    }
#pragma unroll
    for (int off = 16; off > 0; off >>= 1) {
        sP += __shfl_xor(sP, off, 32);
        sT += __shfl_xor(sT, off, 32);
    }
    if (lane == 0) { redA[wav] = sP; redB[wav] = sT; }
    __syncthreads();
    float SP = 0.0f, ST = 0.0f;
#pragma unroll
    for (int w = 0; w < NWAVES; ++w) { SP += redA[w]; ST += redB[w]; }
    __syncthreads();

    // LSE_p = mP + log SP; LSE_t = mT + log ST.
    // KL term: q * ((t - LSE_t) - (p - LSE_p)) = q * ((t - p) + C), C = LSE_p - LSE_t
    const float C     = (mP + __logf(SP)) - (mT + __logf(ST));
    const float invST = __frcp_rn(ST);

    // ---- Pass C: KL accumulation ----
    float kl = 0.0f;
#pragma unroll
    for (int j = 0; j < 16; ++j) {
        bool valid = (t[j] == t[j]);
        float q = et[j] * invST;                   // 0 at masked slots
        float d = valid ? ((t[j] - p[j]) + C) : 0.0f;  // guard NaN * 0
        kl = fmaf(q, d, kl);
    }
#pragma unroll
    for (int off = 16; off > 0; off >>= 1) kl += __shfl_xor(kl, off, 32);
    if (lane == 0) redA[wav] = kl;
    __syncthreads();

    if (tid == 0) {
        float K = 0.0f;
#pragma unroll
        for (int w = 0; w < NWAVES; ++w) K += redA[w];
        bool ok = (cntF > 1.5f);                   // >1 valid entry
        row_kl[row]    = ok ? K : 0.0f;            // gates NaN from empty rows
        row_valid[row] = ok ? 1.0f : 0.0f;
    }
}

__global__ __launch_bounds__(1024) void listnet_finalize_kernel(
    const float* __restrict__ row_kl, const float* __restrict__ row_valid,
    float* __restrict__ out)
{
    __shared__ float sk[32];
    __shared__ float sv[32];
    const int tid  = threadIdx.x;
    const int lane = tid & 31;
    const int wav  = tid >> 5;

    float k = 0.0f, v = 0.0f;
#pragma unroll
    for (int i = tid; i < B_ROWS; i += 1024) { k += row_kl[i]; v += row_valid[i]; }
#pragma unroll
    for (int off = 16; off > 0; off >>= 1) {
        k += __shfl_xor(k, off, 32);
        v += __shfl_xor(v, off, 32);
    }
    if (lane == 0) { sk[wav] = k; sv[wav] = v; }
    __syncthreads();
    if (tid == 0) {
        float K = 0.0f, V = 0.0f;
#pragma unroll
        for (int w = 0; w < 32; ++w) { K += sk[w]; V += sv[w]; }
        out[0] = (V > 0.5f) ? (K / V) : 0.0f;
    }
}

extern "C" void kernel_launch(void* const* d_in, const int* in_sizes, int n_in,
                              void* d_out, int out_size, void* d_ws, size_t ws_size,
                              hipStream_t stream) {
    (void)in_sizes; (void)n_in; (void)out_size; (void)ws_size;
    const float* pred = (const float*)d_in[0];
    const float* targ = (const float*)d_in[1];
    float* out       = (float*)d_out;
    float* row_kl    = (float*)d_ws;            // [B_ROWS]
    float* row_valid = row_kl + B_ROWS;         // [B_ROWS] (64 KB total, fully overwritten)

    listnet_row_kernel<<<B_ROWS, TPB, 0, stream>>>(pred, targ, row_kl, row_valid);
    listnet_finalize_kernel<<<1, 1024, 0, stream>>>(row_kl, row_valid, out);
}